// RNN_cell_38457137169047
// MI455X (gfx1250) — compile-verified
//
#include <hip/hip_runtime.h>
#include <math.h>

typedef __attribute__((ext_vector_type(2))) float v2f;
typedef __attribute__((ext_vector_type(8))) float v8f;
typedef unsigned int u32x4 __attribute__((ext_vector_type(4)));
typedef int i32x8 __attribute__((ext_vector_type(8)));
typedef int i32x4 __attribute__((ext_vector_type(4)));

#if defined(__has_builtin)
#if __has_builtin(__builtin_amdgcn_tensor_load_to_lds)
#define HAVE_TDM 1
#endif
#endif

// ---------------------------------------------------------------------------
// Tunables for the staged recurrent GEMMs: A is [64 x 1024] f32, one block
// owns one 16-row M-tile and 8 N-tiles (8 waves, 256 threads). A chunk of
// 16x256 f32 is DMA'd to LDS by the Tensor Data Mover, double buffered.
// TDM pad: 1 dword pad every 256 dwords -> LDS row stride 257 dwords, so the
// 16 A-fragment lanes hit distinct banks (257 mod 64 == 1).
// ---------------------------------------------------------------------------
#define KTOT 1024
#define KCH 256
#define NCHUNK 4
#define ROWSTRIDE 257              // dwords per staged row (256 + TDM pad)
#define BUFDW (16 * ROWSTRIDE)     // dwords per buffer
#define SHDW (2 * BUFDW)           // double buffered

#ifdef HAVE_TDM
// Build a 2D tensor D# and issue TENSOR_LOAD_TO_LDS (6-arg builtin:
// groups 0..3 + trailing group, zero-filled beyond group 1 for a 2D tile).
// Tile: 16 rows x 256 f32, row stride 1024 f32, pad 1 dword / 256 dwords.
__device__ __forceinline__ void tdm_issue(unsigned long long gaddr,
                                          unsigned int lds_addr) {
  u32x4 g0;
  g0[0] = 1u;                                       // count=1, user D#
  g0[1] = lds_addr;                                 // LDS byte address
  g0[2] = (unsigned int)gaddr;                      // global_addr[31:0]
  g0[3] = (unsigned int)(gaddr >> 32) | 0x80000000u; // addr[56:32] | type=2
  i32x8 g1;
  g1[0] = (int)((2u << 16) | (1u << 20) | (7u << 22)); // 4B elems, pad_en,
                                                       // interval=256dw, amt=1dw
  g1[1] = (int)(256u << 16);  // tensor_dim0 = 256 (bits 79:48, low half)
  g1[2] = (int)(16u << 16);   // tensor_dim0 hi | tensor_dim1 = 16
  g1[3] = (int)(256u << 16);  // tensor_dim1 hi | tile_dim0 = 256
  g1[4] = (int)16u;           // tile_dim1 = 16, tile_dim2 = 0
  g1[5] = (int)1024u;         // tensor_dim0_stride = 1024 elems (low 32)
  g1[6] = 0;                  // stride0 hi | tensor_dim1_stride lo
  g1[7] = 0;
  i32x4 z4 = {0, 0, 0, 0};
  i32x8 z8 = {0, 0, 0, 0, 0, 0, 0, 0};
  __builtin_amdgcn_tensor_load_to_lds(g0, g1, z4, z4, z8, 0);
}
#endif

// ---------------------------------------------------------------------------
// Wave-level 16x16 tile dot product, A fragment from LDS, B from global.
// Shared-A staged GEMM core: executed by all 256 threads of the block.
// Returns the v8f accumulator for this wave's 16x16 tile.
// ---------------------------------------------------------------------------
__device__ __forceinline__ v8f staged_gemm_f32(const float* __restrict__ A,
                                               int tm,
                                               const float* __restrict__ bRow,
                                               float* sh) {
  int lane = threadIdx.x & 31, l = lane & 15, half = lane >> 4;
  bool issuer = (threadIdx.x < 32);   // wave 0 drives the TDM
  (void)issuer;
  unsigned long long gbase =
      (unsigned long long)(uintptr_t)(A + (size_t)tm * 16 * KTOT);
  unsigned int ldsbase = (unsigned int)(uintptr_t)sh;
  (void)gbase; (void)ldsbase;
  v8f acc = {};
#ifdef HAVE_TDM
  if (issuer) tdm_issue(gbase, ldsbase);
#endif
  for (int c = 0; c < NCHUNK; ++c) {
    float* buf = sh + ((c & 1) ? BUFDW : 0);
#ifdef HAVE_TDM
    if (issuer) __builtin_amdgcn_s_wait_tensorcnt(0);  // chunk c landed in LDS
    __syncthreads();                                   // publish to all waves
    if (issuer && c + 1 < NCHUNK)                      // prefetch next chunk
      tdm_issue(gbase + (unsigned long long)(c + 1) * KCH * 4,
                ldsbase + (unsigned int)(((c + 1) & 1) * BUFDW * 4));
#else
    __syncthreads();
    for (int i = threadIdx.x; i < 16 * KCH; i += 256) {
      int r = i >> 8, kk = i & (KCH - 1);
      buf[r * ROWSTRIDE + kk] =
          A[(size_t)(tm * 16 + r) * KTOT + c * KCH + kk];
    }
    __syncthreads();
#endif
    const float* as = buf + l * ROWSTRIDE + 2 * half;
    const float* bs = bRow + c * KCH;
#pragma unroll 8
    for (int k = 0; k < KCH; k += 4) {
      v2f a = *(const v2f*)(as + k);          // ds_load_b64, bank-spread
      v2f b = *(const v2f*)(bs + k);          // global_load_b64 (L2-resident)
      acc = __builtin_amdgcn_wmma_f32_16x16x4_f32(
          false, a, false, b, (short)0, acc, false, false);
    }
    __syncthreads();                          // buffer reusable at c+2
  }
  return acc;
}

__device__ __forceinline__ float activate(float x, int act) {
  if (act == 0) return tanhf(x);
  if (act == 1) return fmaxf(x, 0.0f);
  if (act == 2) return 1.0f / (1.0f + __expf(-x));
  return x;
}

// ---------------------------------------------------------------------------
// Staged GEMM (node 1): C = act( A @ Bt^T + bias + addend ), N = 1024.
// grid = (4 m-tiles * 8 n-groups), block = 256 (8 waves, 8 n-tiles).
// ---------------------------------------------------------------------------
__global__ void kgemm_staged(const float* __restrict__ A,
                             const float* __restrict__ Bt,
                             const float* __restrict__ bias,
                             const float* __restrict__ addend,
                             float* __restrict__ C, int act) {
  __shared__ float sh[SHDW];
  const int N = 1024;
  int w = threadIdx.x >> 5, lane = threadIdx.x & 31;
  int l = lane & 15, half = lane >> 4;
  int tm = blockIdx.x >> 3;
  int tn = ((blockIdx.x & 7) << 3) + w;
  const float* bRow = Bt + (size_t)(tn * 16 + l) * KTOT + 2 * half;
  v8f acc = staged_gemm_f32(A, tm, bRow, sh);
  int n = tn * 16 + l;
  int m0 = tm * 16 + half * 8;
  float bn = bias[n];
#pragma unroll
  for (int r = 0; r < 8; ++r) {
    size_t idx = (size_t)(m0 + r) * N + n;
    float v = acc[r] + bn;
    if (addend) v += addend[idx];
    C[idx] = activate(v, act);
  }
}

// ---------------------------------------------------------------------------
// Staged sibling pair (nodes 2+3 / 4+5): blockIdx.y selects the weight set.
// ---------------------------------------------------------------------------
__global__ void kgemm_pair_staged(const float* __restrict__ A,
                                  const float* __restrict__ Bt0,
                                  const float* __restrict__ bias0, int act0,
                                  float* __restrict__ C0,
                                  const float* __restrict__ Bt1,
                                  const float* __restrict__ bias1, int act1,
                                  float* __restrict__ C1) {
  __shared__ float sh[SHDW];
  const int N = 1024;
  const float* Bt = blockIdx.y ? Bt1 : Bt0;
  const float* bias = blockIdx.y ? bias1 : bias0;
  float* C = blockIdx.y ? C1 : C0;
  int act = blockIdx.y ? act1 : act0;
  int w = threadIdx.x >> 5, lane = threadIdx.x & 31;
  int l = lane & 15, half = lane >> 4;
  int tm = blockIdx.x >> 3;
  int tn = ((blockIdx.x & 7) << 3) + w;
  const float* bRow = Bt + (size_t)(tn * 16 + l) * KTOT + 2 * half;
  v8f acc = staged_gemm_f32(A, tm, bRow, sh);
  int n = tn * 16 + l;
  int m0 = tm * 16 + half * 8;
  float bn = bias[n];
#pragma unroll
  for (int r = 0; r < 8; ++r) {
    size_t idx = (size_t)(m0 + r) * N + n;
    C[idx] = activate(acc[r] + bn, act);
  }
}

// ---------------------------------------------------------------------------
// Node 6 staged GEMM (tanh) fused with the softmax-weighted leaf combine:
// h_new = cw0*L0 + cw1*L1 + cw2*tanh(A @ Bt^T + bias)
// ---------------------------------------------------------------------------
__global__ void kgemm_leaf_staged(const float* __restrict__ A,
                                  const float* __restrict__ Bt,
                                  const float* __restrict__ bias,
                                  const float* __restrict__ L0,
                                  const float* __restrict__ L1,
                                  const float* __restrict__ cw,
                                  float* __restrict__ Hout) {
  __shared__ float sh[SHDW];
  const int N = 1024;
  int w = threadIdx.x >> 5, lane = threadIdx.x & 31;
  int l = lane & 15, half = lane >> 4;
  int tm = blockIdx.x >> 3;
  int tn = ((blockIdx.x & 7) << 3) + w;
  const float* bRow = Bt + (size_t)(tn * 16 + l) * KTOT + 2 * half;
  v8f acc = staged_gemm_f32(A, tm, bRow, sh);
  int n = tn * 16 + l;
  int m0 = tm * 16 + half * 8;
  float bn = bias[n];
  float c0 = cw[0], c1 = cw[1], c2 = cw[2];
#pragma unroll
  for (int r = 0; r < 8; ++r) {
    size_t idx = (size_t)(m0 + r) * N + n;
    float o6 = tanhf(acc[r] + bn);
    Hout[idx] = c0 * L0[idx] + c1 * L1[idx] + c2 * o6;
  }
}

// ---------------------------------------------------------------------------
// Simple per-wave GEMM (used for the HBM-bound final W_out GEMM):
// C[m][n] = act( A @ Bt^T + bias ), one wave per 16x16 tile.
// ---------------------------------------------------------------------------
__global__ void kgemm(const float* __restrict__ A,
                      const float* __restrict__ Bt,
                      const float* __restrict__ bias,
                      float* __restrict__ C,
                      int N, int K, int tilesN, int act) {
  int lane = threadIdx.x, l = lane & 15, half = lane >> 4;
  int tm = blockIdx.x / tilesN, tn = blockIdx.x % tilesN;
  const float* aPtr = A + (size_t)(tm * 16 + l) * K + 2 * half;
  const float* bPtr = Bt + (size_t)(tn * 16 + l) * K + 2 * half;
  v8f acc = {};
#pragma unroll 8
  for (int k = 0; k < K; k += 4) {
    v2f a = *(const v2f*)(aPtr + k);
    v2f b = *(const v2f*)(bPtr + k);
    acc = __builtin_amdgcn_wmma_f32_16x16x4_f32(
        false, a, false, b, (short)0, acc, false, false);
  }
  int n = tn * 16 + l;
  int m0 = tm * 16 + half * 8;
  float bn = bias ? bias[n] : 0.0f;
#pragma unroll
  for (int r = 0; r < 8; ++r) {
    size_t idx = (size_t)(m0 + r) * N + n;
    C[idx] = activate(acc[r] + bn, act);
  }
}

// ---------------------------------------------------------------------------
// Embedding gather + node-0 GEMM: Xe[m][n] = word2vec[tok(m)] @ W0^T + b0.
// ---------------------------------------------------------------------------
__global__ void kembed(const float* __restrict__ word2vec,
                       const int* __restrict__ tokens, // [64,128] row-major
                       const float* __restrict__ W0,   // [1024,1024]
                       const float* __restrict__ b0,
                       float* __restrict__ Xe,
                       int rowBase, int tilesN, int K) {
  int lane = threadIdx.x, l = lane & 15, half = lane >> 4;
  int tm = blockIdx.x / tilesN, tn = blockIdx.x % tilesN;
  int grow = rowBase + tm * 16 + l;
  int t = grow >> 6, b = grow & 63;
  int tok = tokens[b * 128 + t + 1];        // scan uses columns 1..T-1
  const float* aPtr = word2vec + (size_t)tok * K + 2 * half;
  const float* bPtr = W0 + (size_t)(tn * 16 + l) * K + 2 * half;
  v8f acc = {};
#pragma unroll 8
  for (int k = 0; k < K; k += 4) {
    v2f a = *(const v2f*)(aPtr + k);
    v2f b = *(const v2f*)(bPtr + k);
    acc = __builtin_amdgcn_wmma_f32_16x16x4_f32(
        false, a, false, b, (short)0, acc, false, false);
  }
  int n = tn * 16 + l;
  int m0 = tm * 16 + half * 8;
  float bn = b0[n];
#pragma unroll
  for (int r = 0; r < 8; ++r)
    Xe[(size_t)(m0 + r) * 1024 + n] = acc[r] + bn;
}

// ---------------------------------------------------------------------------
// clean_w = softmax( (W_sum @ actions + b_sum)[ {1,3,5} ] )
// ---------------------------------------------------------------------------
__global__ void kweights(const float* __restrict__ W_sum,
                         const float* __restrict__ b_sum,
                         const float* __restrict__ actions,
                         float* __restrict__ cw) {
  if (threadIdx.x == 0) {
    const int leaf[3] = {1, 3, 5};
    float w[3], mx = -3.4e38f;
    for (int j = 0; j < 3; ++j) {
      float s = b_sum[leaf[j]];
      for (int a = 0; a < 12; ++a) s += W_sum[leaf[j] * 12 + a] * actions[a];
      w[j] = s;
      mx = fmaxf(mx, s);
    }
    float den = 0.0f;
    for (int j = 0; j < 3; ++j) { w[j] = __expf(w[j] - mx); den += w[j]; }
    for (int j = 0; j < 3; ++j) cw[j] = w[j] / den;
  }
}

// ---------------------------------------------------------------------------
extern "C" void kernel_launch(void* const* d_in, const int* in_sizes, int n_in,
                              void* d_out, int out_size, void* d_ws,
                              size_t ws_size, hipStream_t stream) {
  (void)in_sizes; (void)n_in; (void)out_size;
  const float* word2vec = (const float*)d_in[0];
  const float* h_in     = (const float*)d_in[1];
  const int*   tokens   = (const int*)d_in[2];
  const float* W0       = (const float*)d_in[3];
  const float* b0       = (const float*)d_in[4];
  const float* Ws       = (const float*)d_in[5];
  const float* bs       = (const float*)d_in[6];
  const float* W_out    = (const float*)d_in[7];
  const float* b_out    = (const float*)d_in[8];
  const float* W_sum    = (const float*)d_in[9];
  const float* b_sum    = (const float*)d_in[10];
  const float* actions  = (const float*)d_in[11];
  float* out = (float*)d_out;

  const int H = 1024, F = 1024, NC = 50000;
  const int steps = 127;
  const size_t BH = (size_t)64 * H;
  const size_t XE_full = (size_t)steps * BH;

  float* p  = (float*)d_ws;
  float* cw = p;  p += 16;
  bool full = ws_size >= (16 + XE_full + 7 * BH) * sizeof(float);
  float* Xe = p;  p += full ? XE_full : BH;
  float* C1 = p;  p += BH;
  float* C3 = p;  p += BH;
  float* C5 = p;  p += BH;
  float* L0 = p;  p += BH;
  float* L1 = p;  p += BH;
  float* h0 = p;  p += BH;
  float* h1 = p;

  kweights<<<1, 32, 0, stream>>>(W_sum, b_sum, actions, cw);

  if (full) {
    // Batched embed GEMM for all 127 steps: M=8128 -> 508 x 64 tiles.
    kembed<<<dim3(508 * 64), 32, 0, stream>>>(word2vec, tokens, W0, b0, Xe,
                                              0, 64, F);
  }

  const float* hsrc = h_in;
  float* hdst = h0;
  for (int t = 0; t < steps; ++t) {
    const float* XeT = full ? (Xe + (size_t)t * BH) : Xe;
    if (!full) {
      kembed<<<dim3(4 * 64), 32, 0, stream>>>(word2vec, tokens, W0, b0, Xe,
                                              t * 64, 64, F);
    }
    // node 1: tanh(h @ Ws0^T + bs0 + Xe_t)   (TDM-staged A)
    kgemm_staged<<<dim3(32), 256, 0, stream>>>(
        hsrc, Ws + 0ull * H * H, bs + 0 * H, XeT, C1, /*tanh*/ 0);
    // nodes 2 (relu -> L0) and 3 (sigmoid -> C3)
    kgemm_pair_staged<<<dim3(32, 2), 256, 0, stream>>>(
        C1, Ws + 1ull * H * H, bs + 1 * H, /*relu*/ 1, L0,
        Ws + 2ull * H * H, bs + 2 * H, /*sigmoid*/ 2, C3);
    // nodes 4 (tanh -> L1) and 5 (relu -> C5)
    kgemm_pair_staged<<<dim3(32, 2), 256, 0, stream>>>(
        C3, Ws + 3ull * H * H, bs + 3 * H, /*tanh*/ 0, L1,
        Ws + 4ull * H * H, bs + 4 * H, /*relu*/ 1, C5);
    // node 6 (tanh leaf) fused with weighted combine -> h_new
    kgemm_leaf_staged<<<dim3(32), 256, 0, stream>>>(
        C5, Ws + 5ull * H * H, bs + 5 * H, L0, L1, cw, hdst);
    hsrc = hdst;
    hdst = (hdst == h0) ? h1 : h0;
  }

  // logits = h @ W_out^T + b_out : M=64, N=50000, K=1024.
  // HBM-bound (200 MB of W_out read once) -> simple per-wave WMMA kernel.
  kgemm<<<dim3(4 * 3125), 32, 0, stream>>>(hsrc, W_out, b_out, out,
                                           NC, H, 3125, /*none*/ -1);
}